// FocalModulation_55817394978821
// MI455X (gfx1250) — compile-verified
//
#include <hip/hip_runtime.h>
#include <hip/hip_bf16.h>
#include <math.h>

// ---------------------------------------------------------------------------
// FocalModulation forward for MI455X (gfx1250), fp32 end-to-end.
// GEMMs use V_WMMA_F32_16X16X4_F32 (wave32 WMMA, D = A*B + C).
// LDS layouts chosen so every WMMA fragment is one aligned ds_load_b64.
// ---------------------------------------------------------------------------

typedef __attribute__((ext_vector_type(2))) float v2f;
typedef __attribute__((ext_vector_type(8))) float v8f;

#define BB   16
#define HH   96
#define WW   96
#define CC   128
#define NPIX (BB * HH * WW)      // 147456
#define HW   (HH * WW)           // 9216
#define OUTF 260
#define LN_EPS 1e-3f
#define ASTRIDE 36               // padded A row stride (8B/16B aligned, bank-clean)

__device__ __forceinline__ float gelu_exact(float v) {
    return 0.5f * v * (1.0f + erff(v * 0.70710678118654752f));
}

// ---------------------------------------------------------------------------
// K1: y = x @ Wf + bf   -> q [N,128], ctx [N,128], gates [N,4]
// Block = 256 threads (8 waves), 64 rows per block, K staged in 4 chunks of 32.
// Wave w: M-tile = w>>1 (4 M-tiles of 16 rows), N-tiles (w&1)*8 .. +7 (256 cols).
// B is stored K-pair packed: (k,col) -> Bs2[k>>1][col*2 + (k&1)], so a WMMA
// B fragment is a single aligned 8-byte LDS read.
// ---------------------------------------------------------------------------
__global__ __launch_bounds__(256) void k1_proj(
    const float* __restrict__ x, const float* __restrict__ Wf,
    const float* __restrict__ bf, float* __restrict__ q,
    float* __restrict__ ctx, float* __restrict__ gates)
{
    __shared__ float As[64][ASTRIDE];
    __shared__ float Bs2[16][512];   // 32 k x 256 cols, pair-packed

    const int t    = threadIdx.x;
    const int wave = t >> 5;
    const int lane = t & 31;
    const int l15  = lane & 15;
    const int hi   = lane >> 4;
    const int r0   = blockIdx.x * 64;
    const int mt   = wave >> 1;
    const int ntb  = (wave & 1) * 8;

    __builtin_amdgcn_s_wait_tensorcnt(0);

    v8f zero = {0.f, 0.f, 0.f, 0.f, 0.f, 0.f, 0.f, 0.f};
    v8f acc[8];
    #pragma unroll
    for (int n = 0; n < 8; ++n) acc[n] = zero;

    float gacc = 0.0f;
    const int grow_g = t >> 2;    // local row 0..63 for gate columns
    const int gcol   = t & 3;     // gate column 0..3

    const float* arow_a = &As[mt * 16 + l15][0];

    for (int kc = 0; kc < 4; ++kc) {
        const int k0 = kc * 32;
        // stage A: 64 rows x 32 k, float4 loads + 16B-aligned LDS stores
        #pragma unroll
        for (int i = 0; i < 2; ++i) {
            int idx = t + i * 256;           // vec4 index over 512
            int row = idx >> 3, kk = (idx & 7) * 4;
            float4 v = *reinterpret_cast<const float4*>(
                &x[(size_t)(r0 + row) * CC + k0 + kk]);
            *reinterpret_cast<float4*>(&As[row][kk]) = v;
        }
        // stage B: 32 k x 256 cols of Wf, float4 loads, pair-packed stores
        #pragma unroll
        for (int i = 0; i < 8; ++i) {
            int idx = t + i * 256;           // vec4 index over 2048
            int kr  = idx >> 6, col = (idx & 63) * 4;
            float4 v = *reinterpret_cast<const float4*>(
                &Wf[(size_t)(k0 + kr) * OUTF + col]);
            float* dst = &Bs2[kr >> 1][col * 2 + (kr & 1)];
            dst[0] = v.x; dst[2] = v.y; dst[4] = v.z; dst[6] = v.w;
        }
        if (kc < 3)  // prefetch next A chunk -> global_prefetch_b8
            __builtin_prefetch(&x[(size_t)(r0 + (t >> 2)) * CC + k0 + 32 + (t & 3) * 8], 0, 1);
        __syncthreads();

        #pragma unroll
        for (int ks = 0; ks < 8; ++ks) {
            const int kb = ks * 4 + hi * 2;              // even K index
            v2f a = *reinterpret_cast<const v2f*>(&arow_a[kb]);
            const float* brow = &Bs2[ks * 2 + hi][0];
            #pragma unroll
            for (int n = 0; n < 8; ++n) {
                v2f b = *reinterpret_cast<const v2f*>(&brow[((ntb + n) * 16 + l15) * 2]);
                acc[n] = __builtin_amdgcn_wmma_f32_16x16x4_f32(
                    false, a, false, b, (short)0, acc[n], false, false);
            }
        }
        // gate columns (256..259) via plain VALU dot product
        {
            const float* wg = Wf + (size_t)k0 * OUTF + 256 + gcol;
            #pragma unroll 8
            for (int kk = 0; kk < 32; ++kk)
                gacc += As[grow_g][kk] * wg[(size_t)kk * OUTF];
        }
        __syncthreads();
    }

    // epilogue: split columns into q (0..127) / ctx (128..255), add bias
    #pragma unroll
    for (int n = 0; n < 8; ++n) {
        const int col  = (ntb + n) * 16 + l15;
        const float bi = bf[col];
        float* dst = (col < CC) ? q : ctx;
        const int ocol = (col < CC) ? col : (col - CC);
        #pragma unroll
        for (int r = 0; r < 8; ++r) {
            const int row = r0 + mt * 16 + hi * 8 + r;
            dst[(size_t)row * CC + ocol] = acc[n][r] + bi;
        }
    }
    gates[(size_t)(r0 + grow_g) * 4 + gcol] = gacc + bf[256 + gcol];
}

// ---------------------------------------------------------------------------
// K2: depthwise conv (SAME, zero pad) + exact GELU + gated accumulation.
// 256 threads = 2 pixels; thread handles one channel of one pixel.
// ---------------------------------------------------------------------------
__global__ __launch_bounds__(256) void dwconv_gate(
    const float* __restrict__ cin, const float* __restrict__ kern,
    const float* __restrict__ gates, float* __restrict__ cout,
    float* __restrict__ call, int ksize, int level, int first)
{
    const int t   = threadIdx.x;
    const int pix = blockIdx.x * 2 + (t >> 7);
    const int c   = t & (CC - 1);
    const int b   = pix / HW;
    const int rem = pix - b * HW;
    const int h   = rem / WW;
    const int w   = rem - h * WW;
    const int r   = ksize >> 1;

    float acc = 0.0f;
    for (int kh = 0; kh < ksize; ++kh) {
        const int hh = h + kh - r;
        if ((unsigned)hh >= (unsigned)HH) continue;
        for (int kw = 0; kw < ksize; ++kw) {
            const int w2 = w + kw - r;
            if ((unsigned)w2 >= (unsigned)WW) continue;
            acc += cin[(((size_t)b * HH + hh) * WW + w2) * CC + c] *
                   kern[(size_t)(kh * ksize + kw) * CC + c];
        }
    }
    const float v = gelu_exact(acc);
    cout[(size_t)pix * CC + c] = v;
    const float g = gates[(size_t)pix * 4 + level];
    const float prev = first ? 0.0f : call[(size_t)pix * CC + c];
    call[(size_t)pix * CC + c] = prev + v * g;
}

// ---------------------------------------------------------------------------
// K3: ctx_global[b,c] = gelu(mean over H,W of final ctx)
// ---------------------------------------------------------------------------
__global__ __launch_bounds__(256) void k3_globalctx(
    const float* __restrict__ ctx, float* __restrict__ ctxg)
{
    __shared__ float red[256];
    const int b    = blockIdx.x;
    const int t    = threadIdx.x;
    const int c    = t & (CC - 1);
    const int half = t >> 7;

    float s = 0.0f;
    const float* p = ctx + (size_t)b * HW * CC;
    for (int pix = half; pix < HW; pix += 2)
        s += p[(size_t)pix * CC + c];
    red[t] = s;
    __syncthreads();
    if (t < CC) {
        const float tot = red[t] + red[t + CC];
        ctxg[b * CC + c] = gelu_exact(tot * (1.0f / (float)HW));
    }
}

// ---------------------------------------------------------------------------
// K4/K6: D = A @ Bm + bias, [N,128]x[128,128].
// fuse=1: A row gets + ctx_global[b]*gate3 during LDS staging (modulator GEMM).
// Wave w: M-tile = w>>1, N-tiles (w&1)*4 .. +3.
// ---------------------------------------------------------------------------
__global__ __launch_bounds__(256) void gemm128_wmma(
    const float* __restrict__ A, const float* __restrict__ Bm,
    const float* __restrict__ bias, float* __restrict__ D,
    const float* __restrict__ ctxg, const float* __restrict__ gates, int fuse)
{
    __shared__ float As[64][ASTRIDE];
    __shared__ float Bs2[16][256];   // 32 k x 128 cols, pair-packed

    const int t    = threadIdx.x;
    const int wave = t >> 5;
    const int lane = t & 31;
    const int l15  = lane & 15;
    const int hi   = lane >> 4;
    const int r0   = blockIdx.x * 64;
    const int mt   = wave >> 1;
    const int ntb  = (wave & 1) * 4;

    __builtin_amdgcn_s_wait_tensorcnt(0);

    v8f zero = {0.f, 0.f, 0.f, 0.f, 0.f, 0.f, 0.f, 0.f};
    v8f acc[4];
    #pragma unroll
    for (int n = 0; n < 4; ++n) acc[n] = zero;

    const float* arow_a = &As[mt * 16 + l15][0];

    for (int kc = 0; kc < 4; ++kc) {
        const int k0 = kc * 32;
        // stage A (vec4), optionally fusing ctx_global*gate3
        #pragma unroll
        for (int i = 0; i < 2; ++i) {
            int idx  = t + i * 256;          // vec4 index over 512
            int row  = idx >> 3, kk = (idx & 7) * 4;
            int grow = r0 + row;
            float4 v = *reinterpret_cast<const float4*>(
                &A[(size_t)grow * CC + k0 + kk]);
            if (fuse) {
                int bidx = grow / HW;
                float g  = gates[(size_t)grow * 4 + 3];
                float4 cg = *reinterpret_cast<const float4*>(
                    &ctxg[bidx * CC + k0 + kk]);
                v.x += cg.x * g; v.y += cg.y * g; v.z += cg.z * g; v.w += cg.w * g;
            }
            *reinterpret_cast<float4*>(&As[row][kk]) = v;
        }
        // stage B (vec4 loads, pair-packed stores)
        #pragma unroll
        for (int i = 0; i < 4; ++i) {
            int idx = t + i * 256;           // vec4 index over 1024
            int kr  = idx >> 5, col = (idx & 31) * 4;
            float4 v = *reinterpret_cast<const float4*>(
                &Bm[(size_t)(k0 + kr) * CC + col]);
            float* dst = &Bs2[kr >> 1][col * 2 + (kr & 1)];
            dst[0] = v.x; dst[2] = v.y; dst[4] = v.z; dst[6] = v.w;
        }
        if (kc < 3)
            __builtin_prefetch(&A[(size_t)(r0 + (t >> 2)) * CC + k0 + 32 + (t & 3) * 8], 0, 1);
        __syncthreads();

        #pragma unroll
        for (int ks = 0; ks < 8; ++ks) {
            const int kb = ks * 4 + hi * 2;
            v2f a = *reinterpret_cast<const v2f*>(&arow_a[kb]);
            const float* brow = &Bs2[ks * 2 + hi][0];
            #pragma unroll
            for (int n = 0; n < 4; ++n) {
                v2f b = *reinterpret_cast<const v2f*>(&brow[((ntb + n) * 16 + l15) * 2]);
                acc[n] = __builtin_amdgcn_wmma_f32_16x16x4_f32(
                    false, a, false, b, (short)0, acc[n], false, false);
            }
        }
        __syncthreads();
    }

    #pragma unroll
    for (int n = 0; n < 4; ++n) {
        const int col  = (ntb + n) * 16 + l15;
        const float bi = bias[col];
        #pragma unroll
        for (int r = 0; r < 8; ++r) {
            const int row = r0 + mt * 16 + hi * 8 + r;
            D[(size_t)row * CC + col] = acc[n][r] + bi;
        }
    }
}

// ---------------------------------------------------------------------------
// K5: x_out = q * modulator, then LayerNorm over C=128 (in-place safe).
// One block (128 threads, 4 waves) per token.
// ---------------------------------------------------------------------------
__global__ __launch_bounds__(128) void k4b_ln(
    const float* __restrict__ q, const float* __restrict__ mod,
    const float* __restrict__ gamma, const float* __restrict__ beta,
    float* __restrict__ xn)
{
    __shared__ float r1[128];
    __shared__ float r2[128];
    const int row = blockIdx.x;
    const int c   = threadIdx.x;

    const float v = q[(size_t)row * CC + c] * mod[(size_t)row * CC + c];
    r1[c] = v;
    r2[c] = v * v;
    __syncthreads();
    #pragma unroll
    for (int s = 64; s > 0; s >>= 1) {
        if (c < s) { r1[c] += r1[c + s]; r2[c] += r2[c + s]; }
        __syncthreads();
    }
    const float mu  = r1[0] * (1.0f / (float)CC);
    const float var = r2[0] * (1.0f / (float)CC) - mu * mu;
    xn[(size_t)row * CC + c] = (v - mu) * rsqrtf(var + LN_EPS) * gamma[c] + beta[c];
}

// ---------------------------------------------------------------------------
// Host launcher. d_in order (setup_inputs dict):
// 0:x 1:Wf 2:bf 3:Wh 4:bh 5:gamma 6:beta 7:Wp 8:bp 9:k0 10:k1 11:k2
// ---------------------------------------------------------------------------
extern "C" void kernel_launch(void* const* d_in, const int* in_sizes, int n_in,
                              void* d_out, int out_size, void* d_ws, size_t ws_size,
                              hipStream_t stream)
{
    const float* x     = (const float*)d_in[0];
    const float* Wf    = (const float*)d_in[1];
    const float* bf    = (const float*)d_in[2];
    const float* Wh    = (const float*)d_in[3];
    const float* bh    = (const float*)d_in[4];
    const float* gamma = (const float*)d_in[5];
    const float* beta  = (const float*)d_in[6];
    const float* Wp    = (const float*)d_in[7];
    const float* bp    = (const float*)d_in[8];
    const float* k0    = (const float*)d_in[9];
    const float* k1    = (const float*)d_in[10];
    const float* k2    = (const float*)d_in[11];

    float* ws = (float*)d_ws;
    const size_t nc = (size_t)NPIX * CC;     // 18,874,368 floats per buffer
    float* q      = ws;                      // [N,128]
    float* ctxA   = ws + nc;                 // ping  (also reused as mod/xn)
    float* ctxB   = ws + 2 * nc;             // pong
    float* ctxall = ws + 3 * nc;             // [N,128]
    float* gates  = ws + 4 * nc;             // [N,4]
    float* ctxg   = gates + (size_t)NPIX * 4;// [B,128]

    const int gemm_blocks = NPIX / 64;       // 2304

    // 1) projection GEMM -> q, ctx(A), gates
    k1_proj<<<gemm_blocks, 256, 0, stream>>>(x, Wf, bf, q, ctxA, gates);
    // 2) focal levels (ping-pong ctx), gated accumulation into ctxall
    dwconv_gate<<<NPIX / 2, 256, 0, stream>>>(ctxA, k0, gates, ctxB, ctxall, 3, 0, 1);
    dwconv_gate<<<NPIX / 2, 256, 0, stream>>>(ctxB, k1, gates, ctxA, ctxall, 5, 1, 0);
    dwconv_gate<<<NPIX / 2, 256, 0, stream>>>(ctxA, k2, gates, ctxB, ctxall, 7, 2, 0);
    // 3) global context from final ctx (in ctxB)
    k3_globalctx<<<BB, 256, 0, stream>>>(ctxB, ctxg);
    // 4) modulator GEMM, ctx_global*gate3 fused into A staging; result -> ctxA
    gemm128_wmma<<<gemm_blocks, 256, 0, stream>>>(ctxall, Wh, bh, ctxA, ctxg, gates, 1);
    // 5) q * modulator + LayerNorm (in-place in ctxA)
    k4b_ln<<<NPIX, 128, 0, stream>>>(q, ctxA, gamma, beta, ctxA);
    // 6) output projection GEMM -> d_out
    gemm128_wmma<<<gemm_blocks, 256, 0, stream>>>(ctxA, Wp, bp, (float*)d_out,
                                                  nullptr, nullptr, 0);
}